// MolGNN_12738873000425
// MI455X (gfx1250) — compile-verified
//
#include <hip/hip_runtime.h>

#define NNODES 10000
#define MPAD   10016          // NNODES padded up to a multiple of BM (32)
#define NEDGES 160000
#define NGRAPH 512
#define FIN    128
#define FH     512
#define FOUT   16
#define NLAYERS 4

typedef __attribute__((ext_vector_type(2))) float v2f;
typedef __attribute__((ext_vector_type(8))) float v8f;

// ------------- CDNA5 async global->LDS copy (ASYNCcnt-tracked) -------------
__device__ __forceinline__ void async_copy_b128(const float* gptr, float* lptr) {
    unsigned lds_off = (unsigned)(uintptr_t)lptr;   // low 32 bits of flat LDS addr = LDS offset
    asm volatile("global_load_async_to_lds_b128 %0, %1, off"
                 :: "v"(lds_off), "v"(gptr) : "memory");
}
__device__ __forceinline__ void wait_async0() {
    asm volatile("s_wait_asynccnt 0x0" ::: "memory");
}

// ---------------- utility fills ----------------
__global__ void fill_kernel(float* __restrict__ p, float v, int n) {
    int i = blockIdx.x * blockDim.x + threadIdx.x;
    if (i < n) p[i] = v;
}

// ---------------- degree / dinv ----------------
__global__ void deg_kernel(const int* __restrict__ dst, float* __restrict__ deg) {
    int e = blockIdx.x * blockDim.x + threadIdx.x;
    if (e < NEDGES) atomicAdd(&deg[dst[e]], 1.0f);
}

__global__ void dinv_kernel(float* __restrict__ deg) {
    int i = blockIdx.x * blockDim.x + threadIdx.x;
    if (i < NNODES) deg[i] = rsqrtf(deg[i] + 1.0f);
}

// ---------------- LayerNorm: one wave32 per node ----------------
__global__ void ln_kernel(const float* __restrict__ x, const float* __restrict__ g,
                          const float* __restrict__ b, float* __restrict__ y, int F) {
    int wid  = (blockIdx.x * blockDim.x + threadIdx.x) >> 5;
    int lane = threadIdx.x & 31;
    if (wid >= NNODES) return;                 // wave-uniform guard
    const float* xr = x + (size_t)wid * F;
    float s = 0.f, ss = 0.f;
    for (int f = lane; f < F; f += 32) { float v = xr[f]; s += v; ss += v * v; }
    for (int o = 16; o > 0; o >>= 1) {
        s  += __shfl_xor(s,  o, 32);
        ss += __shfl_xor(ss, o, 32);
    }
    float inv  = 1.0f / (float)F;
    float mu   = s * inv;
    float var  = ss * inv - mu * mu;
    float rstd = rsqrtf(var + 1e-5f);
    float* yr = y + (size_t)wid * F;
    for (int f = lane; f < F; f += 32)
        yr[f] = (xr[f] - mu) * rstd * g[f] + b[f];
}

// ---------------- fp32 WMMA GEMM with async-LDS double buffering ----------------
// C[MPAD x 512] = A[MPAD x K] @ W[K x 512]
// Block: 256 threads (8 waves). Block tile BM=32 x BN=256. K chunk KB=16.
// Wave w: row-tile rt = w>>2 (16 rows), col group cw = w&3 (64 cols -> 4 accumulators).
// Each A fragment (ds_load_b64) feeds 4 WMMAs; B fragments shared via LDS.
// V_WMMA_F32_16X16X4_F32 lane layout (ISA 7.12.2):
//   A (16x4):  lanes 0-15 -> M=lane, K={0,1}; lanes 16-31 -> M=lane-16, K={2,3}
//   B (4x16):  lanes 0-15 -> N=lane, K={0,1}; lanes 16-31 -> N=lane-16, K={2,3}
//   C/D:       v[r] -> row r (lanes 0-15) / row r+8 (lanes 16-31), col = lane&15
#define BM 32
#define BN 256
#define KB 16

__global__ __launch_bounds__(256)
void gemm_wmma_lds_kernel(const float* __restrict__ A, const float* __restrict__ W,
                          float* __restrict__ C, int K) {
    __shared__ float sA[2][BM * KB];   //  4 KB
    __shared__ float sB[2][KB * BN];   // 32 KB

    const int tid  = threadIdx.x;
    const int lane = tid & 31;
    const int wv   = tid >> 5;         // 0..7
    const int rt   = wv >> 2;          // row tile 0/1
    const int cw   = wv & 3;           // col group 0..3

    const int nbn  = 512 / BN;         // 2 column blocks
    const int bm   = blockIdx.x / nbn;
    const int bn   = blockIdx.x % nbn;
    const int row0 = bm * BM;
    const int col0 = bn * BN;

    // loader assignment
    const int arow = tid >> 2;         // threads 0..127: A rows 0..31, 4 x b128 per row
    const int aseg = tid & 3;

    const int nch = K / KB;
    int buf = 0;

    // prologue: stream chunk 0
    if (tid < 128)
        async_copy_b128(A + (size_t)(row0 + arow) * K + aseg * 4,
                        &sA[0][arow * KB + aseg * 4]);
    #pragma unroll
    for (int r = 0; r < 4; ++r) {
        int idx = r * 256 + tid;       // 0..1023 b128 loads
        int br  = idx >> 6;            // B row 0..15
        int bs  = idx & 63;            // 4-float segment
        async_copy_b128(W + (size_t)br * 512 + col0 + bs * 4,
                        &sB[0][br * BN + bs * 4]);
    }
    wait_async0();
    __syncthreads();

    v8f acc[4] = {};
    const int khalf = (lane >> 4) << 1;   // 0 or 2
    const int lrow  = lane & 15;

    for (int ch = 0; ch < nch; ++ch) {
        if (ch + 1 < nch) {                       // stream next chunk into other buffer
            int k0 = (ch + 1) * KB;
            int nb = buf ^ 1;
            if (tid < 128)
                async_copy_b128(A + (size_t)(row0 + arow) * K + k0 + aseg * 4,
                                &sA[nb][arow * KB + aseg * 4]);
            #pragma unroll
            for (int r = 0; r < 4; ++r) {
                int idx = r * 256 + tid;
                int br  = idx >> 6;
                int bs  = idx & 63;
                async_copy_b128(W + (size_t)(k0 + br) * 512 + col0 + bs * 4,
                                &sB[nb][br * BN + bs * 4]);
            }
        }
        // compute current chunk from LDS
        const float* a_base = &sA[buf][(rt * 16 + lrow) * KB];
        #pragma unroll
        for (int kk = 0; kk < KB; kk += 4) {
            v2f a;
            a.x = a_base[kk + khalf];
            a.y = a_base[kk + khalf + 1];
            const float* b0r = &sB[buf][(kk + khalf) * BN];
            const float* b1r = b0r + BN;
            #pragma unroll
            for (int j = 0; j < 4; ++j) {
                int c = cw * 64 + j * 16 + lrow;
                v2f b; b.x = b0r[c]; b.y = b1r[c];
                acc[j] = __builtin_amdgcn_wmma_f32_16x16x4_f32(false, a, false, b,
                                                               (short)0, acc[j], false, false);
            }
        }
        wait_async0();       // next-chunk async copies complete
        __syncthreads();     // and everyone is done reading current buffer
        buf ^= 1;
    }

    const int orow = row0 + rt * 16 + ((lane >> 4) << 3);
    #pragma unroll
    for (int j = 0; j < 4; ++j) {
        int c = col0 + cw * 64 + j * 16 + lrow;
        #pragma unroll
        for (int r = 0; r < 8; ++r)
            C[(size_t)(orow + r) * 512 + c] = acc[j][r];
    }
}

// ---------------- per-(edge,feature) normalized scatter-add ----------------
__global__ void edge_scatter_kernel(const float* __restrict__ h, const int* __restrict__ src,
                                    const int* __restrict__ dst, const float* __restrict__ dinv,
                                    float* __restrict__ agg) {
    int idx = blockIdx.x * blockDim.x + threadIdx.x;   // over NEDGES*FH (< 2^31)
    if (idx >= NEDGES * FH) return;
    int e = idx >> 9;
    int f = idx & (FH - 1);
    int s = src[e], d = dst[e];
    float c = dinv[s] * dinv[d];
    atomicAdd(&agg[(size_t)d * FH + f], h[(size_t)s * FH + f] * c);
}

// ---------------- self-loop + bias + ReLU (in place on h) ----------------
__global__ void finish_kernel(const float* __restrict__ agg, const float* __restrict__ dinv,
                              const float* __restrict__ bias, float* __restrict__ h) {
    int idx = blockIdx.x * blockDim.x + threadIdx.x;   // over NNODES*FH
    if (idx >= NNODES * FH) return;
    int i = idx >> 9;
    int f = idx & (FH - 1);
    float di = dinv[i];
    float v = agg[idx] + h[idx] * di * di + bias[f];
    h[idx] = v > 0.f ? v : 0.f;
}

// ---------------- pooling ----------------
__global__ void count_kernel(const int* __restrict__ batch, float* __restrict__ counts) {
    int i = blockIdx.x * blockDim.x + threadIdx.x;
    if (i < NNODES) atomicAdd(&counts[batch[i]], 1.0f);
}

__global__ void pool_kernel(const float* __restrict__ h, const int* __restrict__ batch,
                            float* __restrict__ pooled) {
    int idx = blockIdx.x * blockDim.x + threadIdx.x;   // over NNODES*FH
    if (idx >= NNODES * FH) return;
    int i = idx >> 9;
    int f = idx & (FH - 1);
    atomicAdd(&pooled[(size_t)batch[i] * FH + f], h[idx]);
}

// ---------------- final linear: out[G x 16] = (pooled/counts) @ linW + linb ----------------
__global__ void final_kernel(const float* __restrict__ pooled, const float* __restrict__ counts,
                             const float* __restrict__ linW, const float* __restrict__ linb,
                             float* __restrict__ out) {
    int idx = blockIdx.x * blockDim.x + threadIdx.x;
    if (idx >= NGRAPH * FOUT) return;
    int g = idx / FOUT, o = idx % FOUT;
    float cnt = counts[g];
    if (cnt < 1.f) cnt = 1.f;
    const float* pr = pooled + (size_t)g * FH;
    float acc = 0.f;
    for (int f = 0; f < FH; ++f) acc += pr[f] * linW[f * FOUT + o];
    out[idx] = acc / cnt + linb[o];
}

static inline int nblk(long n, int bs) { return (int)((n + bs - 1) / bs); }

extern "C" void kernel_launch(void* const* d_in, const int* in_sizes, int n_in,
                              void* d_out, int out_size, void* d_ws, size_t ws_size,
                              hipStream_t stream) {
    const float* x    = (const float*)d_in[0];
    const int*   eidx = (const int*)  d_in[1];
    const int*   batch= (const int*)  d_in[2];
    const float* W0   = (const float*)d_in[3];
    const float* b0   = (const float*)d_in[4];
    const float* g0   = (const float*)d_in[5];
    const float* be0  = (const float*)d_in[6];
    const float* Wh   = (const float*)d_in[7];
    const float* bh   = (const float*)d_in[8];
    const float* gh   = (const float*)d_in[9];
    const float* beh  = (const float*)d_in[10];
    const float* linW = (const float*)d_in[11];
    const float* linb = (const float*)d_in[12];
    float* out = (float*)d_out;

    const int* src = eidx;            // edge_index[0]
    const int* dst = eidx + NEDGES;   // edge_index[1]

    // workspace layout (floats); GEMM in/out buffers padded to MPAD rows
    float* ws     = (float*)d_ws;
    float* dinv   = ws;                                  // NNODES (padded to 10240)
    float* xn     = ws + 10240;                          // MPAD*FH
    float* hbuf   = xn   + (size_t)MPAD * FH;            // MPAD*FH
    float* agg    = hbuf + (size_t)MPAD * FH;            // NNODES*FH
    float* pooled = agg  + (size_t)NNODES * FH;          // NGRAPH*FH
    float* counts = pooled + (size_t)NGRAPH * FH;        // NGRAPH

    const int NH = NNODES * FH;

    // zero staging buffer once so padded rows are benign, then degree -> dinv
    fill_kernel<<<nblk((long)MPAD * FH,256),256,0,stream>>>(xn, 0.f, MPAD * FH);
    fill_kernel<<<nblk(NNODES,256),256,0,stream>>>(dinv, 0.f, NNODES);
    deg_kernel <<<nblk(NEDGES,256),256,0,stream>>>(dst, dinv);
    dinv_kernel<<<nblk(NNODES,256),256,0,stream>>>(dinv);

    const int gemm_blocks = (MPAD / BM) * (512 / BN);    // 313 * 2 = 626

    for (int l = 0; l < NLAYERS; ++l) {
        const float* lin_in = (l == 0) ? x   : hbuf;
        int          F      = (l == 0) ? FIN : FH;
        const float* g      = (l == 0) ? g0  : gh  + (size_t)(l - 1) * FH;
        const float* be     = (l == 0) ? be0 : beh + (size_t)(l - 1) * FH;
        const float* W      = (l == 0) ? W0  : Wh  + (size_t)(l - 1) * FH * FH;
        const float* b      = (l == 0) ? b0  : bh  + (size_t)(l - 1) * FH;

        ln_kernel<<<nblk((long)NNODES * 32,256),256,0,stream>>>(lin_in, g, be, xn, F);
        gemm_wmma_lds_kernel<<<gemm_blocks,256,0,stream>>>(xn, W, hbuf, F);
        fill_kernel<<<nblk(NH,256),256,0,stream>>>(agg, 0.f, NH);
        edge_scatter_kernel<<<nblk((long)NEDGES * FH,256),256,0,stream>>>(hbuf, src, dst, dinv, agg);
        finish_kernel<<<nblk(NH,256),256,0,stream>>>(agg, dinv, b, hbuf);
    }

    // pooling + final linear
    fill_kernel<<<nblk(NGRAPH * FH,256),256,0,stream>>>(pooled, 0.f, NGRAPH * FH);
    fill_kernel<<<nblk(NGRAPH,256),256,0,stream>>>(counts, 0.f, NGRAPH);
    count_kernel<<<nblk(NNODES,256),256,0,stream>>>(batch, counts);
    pool_kernel<<<nblk((long)NNODES * FH,256),256,0,stream>>>(hbuf, batch, pooled);
    final_kernel<<<nblk(NGRAPH * FOUT,256),256,0,stream>>>(pooled, counts, linW, linb, out);
}